// GLIF_deprecated_69166153335598
// MI455X (gfx1250) — compile-verified
//
#include <hip/hip_runtime.h>

typedef __attribute__((ext_vector_type(2))) float v2f;
typedef __attribute__((ext_vector_type(8))) float v8f;

#define NN 8192
#define KSPLIT 4                 // K-chunks per 16-row tile (deterministic partials, no atomics)
#define KCHUNK (NN / KSPLIT)     // 2048

// ---------------------------------------------------------------------------
// Kernel 1: y_partial[chunk][row] = sum_{k in chunk} W[row,k] * g[k]
// One wave (32 lanes) per (row-tile, k-chunk). 4 waves per block = one tile.
// MACs via V_WMMA_F32_16X16X4_F32; B = g broadcast across all 16 columns.
// ---------------------------------------------------------------------------
__global__ void __launch_bounds__(128)
glif_matvec_wmma(const float* __restrict__ w, const float* __restrict__ g,
                 float* __restrict__ partial) {
    const int lane  = threadIdx.x & 31;
    const int wave  = threadIdx.x >> 5;       // 0..3 -> k-chunk
    const int tile  = blockIdx.x;             // 0..511 -> 16-row tile
    const int row0  = tile * 16;
    const int m     = lane & 15;              // row within tile (A-matrix M)
    const int ksel  = (lane >> 4) << 1;       // lanes 0-15: K=0,1  lanes 16-31: K=2,3
    const int kbeg  = wave * KCHUNK;
    const int kend  = kbeg + KCHUNK;

    const float* wrow = w + (size_t)(row0 + m) * NN;

    v8f c0 = {0.f,0.f,0.f,0.f,0.f,0.f,0.f,0.f};
    v8f c1 = {0.f,0.f,0.f,0.f,0.f,0.f,0.f,0.f};

    for (int k0 = kbeg; k0 < kend; k0 += 32) {
        // stream-prefetch this lane's row ~8KB ahead (speculative; OOB is dropped)
        __builtin_prefetch(wrow + k0 + 2048, 0, 0);
        #pragma unroll
        for (int kk = 0; kk < 32; kk += 8) {
            v2f a0 = *(const v2f*)(wrow + k0 + kk + ksel);
            v2f b0 = *(const v2f*)(g    + k0 + kk + ksel);
            c0 = __builtin_amdgcn_wmma_f32_16x16x4_f32(
                     false, a0, false, b0, (short)0, c0, false, false);
            v2f a1 = *(const v2f*)(wrow + k0 + kk + 4 + ksel);
            v2f b1 = *(const v2f*)(g    + k0 + kk + 4 + ksel);
            c1 = __builtin_amdgcn_wmma_f32_16x16x4_f32(
                     false, a1, false, b1, (short)0, c1, false, false);
        }
    }

    v8f c = c0 + c1;

    // All 16 columns of D are identical. C/D layout: VGPR r -> M=r (lanes 0-15),
    // M=r+8 (lanes 16-31). Lane 0 writes rows 0..7, lane 16 writes rows 8..15.
    if ((lane & 15) == 0) {
        float* dst = partial + (size_t)wave * NN + row0 + ((lane >> 4) << 3);
        #pragma unroll
        for (int r = 0; r < 8; ++r) dst[r] = c[r];
    }
}

// ---------------------------------------------------------------------------
// Kernel 2: elementwise GLIF update.
// I = sigmoid(sum(partials) + x_in); out = [v_new (N) | spiked_soft (N)]
// ---------------------------------------------------------------------------
__global__ void __launch_bounds__(256)
glif_update(const float* __restrict__ partial, const float* __restrict__ x_in,
            const float* __restrict__ v, const float* __restrict__ v_rest,
            const float* __restrict__ tau_m, const float* __restrict__ theta_s,
            const float* __restrict__ theta_v, float* __restrict__ out) {
    int i = blockIdx.x * blockDim.x + threadIdx.x;
    if (i >= NN) return;

    float acc = x_in[i];
    #pragma unroll
    for (int cch = 0; cch < KSPLIT; ++cch) acc += partial[(size_t)cch * NN + i];

    float I   = 1.0f / (1.0f + __expf(-acc));               // sigmoid(w@g + x_in)
    float vr  = v_rest[i];
    float vi  = v[i];
    float vm  = vi + (vr - vi + I) / tau_m[i];              // membrane integration
    float th  = theta_s[i] + theta_v[i];
    float soft = 1.0f / (1.0f + __expf(-(vm - th)));        // surrogate spike
    float spiked = (vm >= th) ? 1.0f : 0.0f;
    float v_new  = spiked * vr + (1.0f - spiked) * vm;      // reset-or-keep

    out[i]      = v_new;
    out[NN + i] = soft;
}

// ---------------------------------------------------------------------------
// Inputs (setup_inputs order): 0:x_in 1:w 2:v 3:g 4:v_rest 5:tau_m 6:tau_g
//                              7:theta_s 8:theta_v 9:b_s 10:a_v 11:b_v
// Output: v_new (N) then spiked_soft (N), both f32.
// ---------------------------------------------------------------------------
extern "C" void kernel_launch(void* const* d_in, const int* in_sizes, int n_in,
                              void* d_out, int out_size, void* d_ws, size_t ws_size,
                              hipStream_t stream) {
    const float* x_in    = (const float*)d_in[0];
    const float* w       = (const float*)d_in[1];
    const float* v       = (const float*)d_in[2];
    const float* g       = (const float*)d_in[3];
    const float* v_rest  = (const float*)d_in[4];
    const float* tau_m   = (const float*)d_in[5];
    const float* theta_s = (const float*)d_in[7];
    const float* theta_v = (const float*)d_in[8];
    float* partial = (float*)d_ws;           // KSPLIT * N floats = 128 KB
    float* out     = (float*)d_out;

    // 512 tiles x 4 waves = 2048 waves streaming W once (256 MB -> ~11 us floor)
    glif_matvec_wmma<<<NN / 16, 128, 0, stream>>>(w, g, partial);
    glif_update<<<(NN + 255) / 256, 256, 0, stream>>>(partial, x_in, v, v_rest,
                                                      tau_m, theta_s, theta_v, out);
}